// LDSMIXTURELQRFNN_58445914964045
// MI455X (gfx1250) — compile-verified
//
#include <hip/hip_runtime.h>
#include <stdint.h>

// ---------------------------------------------------------------------------
// Sequential LDS-mixture-LQR scan, T = 131072 steps.
// Latency-bound serial recurrence -> 1 wave, lane 0 scans; all 32 lanes drive
// a double-buffered async global->LDS input pipeline (CDNA5 ASYNCcnt path).
// ---------------------------------------------------------------------------

#define T_STEPS 131072
#define TILE    256
#define NTILES  (T_STEPS / TILE)
#define B_PRIME (1.0f / 60.0f)

#define AS1 __attribute__((address_space(1)))
#define AS3 __attribute__((address_space(3)))

typedef int v4i __attribute__((vector_size(4 * sizeof(int))));

// Issue one 16B async copy global -> LDS (per active lane).
// Low 32 bits of a flat pointer into LDS are the wave-relative LDS byte
// address.
__device__ __forceinline__ void async_load_b128(const float* g, const float* l) {
  unsigned loff = (unsigned)(uintptr_t)l;
#if __has_builtin(__builtin_amdgcn_global_load_async_to_lds_b128)
  __builtin_amdgcn_global_load_async_to_lds_b128(
      (AS1 v4i*)(uintptr_t)g, (AS3 v4i*)(uintptr_t)loff, 0, 0);
#else
  asm volatile("global_load_async_to_lds_b128 %0, %1, off"
               :
               : "v"(loff), "v"((unsigned long long)(uintptr_t)g)
               : "memory");
#endif
}

template <int N>
__device__ __forceinline__ void wait_async() {
#if __has_builtin(__builtin_amdgcn_s_wait_asynccnt)
  __builtin_amdgcn_s_wait_asynccnt(N);
#else
  asm volatile("s_wait_asynccnt %0" ::"i"(N) : "memory");
#endif
  asm volatile("" ::: "memory");  // compiler barrier: keep DS reads below
}

// Copy one tile (TILE steps of xstar1 + xstar2 = 2 x 2048 B) into LDS buffer
// `buf` via 8 async b128 instructions across the 32 lanes.
__device__ __forceinline__ void issue_tile(const float* xs1g, const float* xs2g,
                                           float (*lin)[2][TILE * 2], int tile,
                                           int buf, int lane) {
  const float* g1 = xs1g + (size_t)tile * (TILE * 2);
  const float* g2 = xs2g + (size_t)tile * (TILE * 2);
  float* l1 = &lin[buf][0][0];
  float* l2 = &lin[buf][1][0];
#pragma unroll
  for (int k = 0; k < 4; ++k) {
    const int off = lane * 4 + k * 128;  // floats: 16B/lane, 512B/instruction
    async_load_b128(g1 + off, l1 + off);
    async_load_b128(g2 + off, l2 + off);
  }
}

__global__ __launch_bounds__(32, 1) void lds_mixture_lqr_scan(
    const float* __restrict__ init_y, const float* __restrict__ xs1g,
    const float* __restrict__ xs2g, const float* __restrict__ Aptr,
    const float* __restrict__ g1p, const float* __restrict__ g2p,
    const float* __restrict__ W1, const float* __restrict__ b1,
    const float* __restrict__ W2, const float* __restrict__ b2,
    float* __restrict__ out) {
  __shared__ float lin[2][2][TILE * 2];  // [buffer][array][TILE*2 floats] = 8KB

  const int lane = threadIdx.x;

  float* out_x = out;                         // (T+1)
  float* out_y = out + (T_STEPS + 1);         // (T+1, 2) row-major
  float* out_u = out + 3 * (T_STEPS + 1);     // (T, 2) row-major

  // Prime the pipeline: tiles 0 and 1 in flight (16 async ops outstanding).
  issue_tile(xs1g, xs2g, lin, 0, 0, lane);
  issue_tile(xs1g, xs2g, lin, 1, 1, lane);

  // Uniform scalar parameters (compiler scalarizes these loads).
  const float a  = Aptr[0];
  const float g1 = g1p[0];
  const float g2 = g2p[0];
  const float w100 = W1[0], w101 = W1[1];
  const float w110 = W1[2], w111 = W1[3];
  const float w120 = W1[4], w121 = W1[5];
  const float w130 = W1[6], w131 = W1[7];
  const float bb0 = b1[0], bb1 = b1[1], bb2 = b1[2], bb3 = b1[3];
  const float v0 = W2[0], v1 = W2[1], v2 = W2[2], v3 = W2[3];
  const float c2 = b2[0];

  float x  = 2.0f;            // X0
  float yx = init_y[0];
  float yy = init_y[1];

  if (lane == 0) {
    out_x[0] = 2.0f;
    out_y[0] = yx;
    out_y[1] = yy;
  }

  for (int tile = 0; tile < NTILES; ++tile) {
    // Tile `tile`'s 8 async loads are the oldest in flight; ASYNCcnt
    // completes in order, so <=8 outstanding means this tile has landed.
    if (tile == NTILES - 1) wait_async<0>(); else wait_async<8>();

    const int buf = tile & 1;
    if (lane == 0) {
      const float2* p1v = (const float2*)&lin[buf][0][0];
      const float2* p2v = (const float2*)&lin[buf][1][0];
      int t = tile * TILE;
#pragma unroll 4
      for (int i = 0; i < TILE; ++i, ++t) {
        const float2 p1 = p1v[i];  // ds_load_b64
        const float2 p2 = p2v[i];

        const float e1x = yx - p1.x, e1y = yy - p1.y;
        const float e2x = yx - p2.x, e2y = yy - p2.y;

        // safe_sigmoid(x) = sigmoid(clip(x, -10, 10))
        const float xc = fminf(10.0f, fmaxf(-10.0f, x));
        const float w  = __builtin_amdgcn_rcpf(1.0f + __expf(-xc));

        const float d1 = __builtin_amdgcn_sqrtf(__builtin_fmaf(e1x, e1x, e1y * e1y));
        const float d2 = __builtin_amdgcn_sqrtf(__builtin_fmaf(e2x, e2x, e2y * e2y));

        // ffn: Linear(2->4) then Linear(4->1), no nonlinearity
        const float h0 = __builtin_fmaf(w100, d1, __builtin_fmaf(w101, d2, bb0));
        const float h1 = __builtin_fmaf(w110, d1, __builtin_fmaf(w111, d2, bb1));
        const float h2 = __builtin_fmaf(w120, d1, __builtin_fmaf(w121, d2, bb2));
        const float h3 = __builtin_fmaf(w130, d1, __builtin_fmaf(w131, d2, bb3));
        const float f  = __builtin_fmaf(
            v0, h0, __builtin_fmaf(v1, h1, __builtin_fmaf(v2, h2, __builtin_fmaf(v3, h3, c2))));

        // u = (-g1*e1)*w + (-g2*e2)*(1-w)
        const float wg1 = g1 * w;
        const float wg2 = g2 - g2 * w;
        const float ux = -__builtin_fmaf(wg1, e1x, wg2 * e2x);
        const float uy = -__builtin_fmaf(wg1, e1y, wg2 * e2y);

        x  = __builtin_fmaf(a, x, f);
        yx = __builtin_fmaf(B_PRIME, ux, yx);
        yy = __builtin_fmaf(B_PRIME, uy, yy);

        // Output bases land on odd-float offsets -> scalar b32 stores.
        out_x[t + 1]           = x;
        out_y[2 * (t + 1)]     = yx;
        out_y[2 * (t + 1) + 1] = yy;
        out_u[2 * t]           = ux;
        out_u[2 * t + 1]       = uy;
      }
    }
    // Buffer `buf` fully consumed -> refill it with tile+2.
    asm volatile("" ::: "memory");
    if (tile + 2 < NTILES) issue_tile(xs1g, xs2g, lin, tile + 2, buf, lane);
  }
}

extern "C" void kernel_launch(void* const* d_in, const int* in_sizes, int n_in,
                              void* d_out, int out_size, void* d_ws, size_t ws_size,
                              hipStream_t stream) {
  (void)in_sizes; (void)n_in; (void)out_size; (void)d_ws; (void)ws_size;
  const float* init_y = (const float*)d_in[0];
  const float* xs1    = (const float*)d_in[1];
  const float* xs2    = (const float*)d_in[2];
  const float* A      = (const float*)d_in[3];
  const float* gain1  = (const float*)d_in[4];
  const float* gain2  = (const float*)d_in[5];
  const float* W1     = (const float*)d_in[6];
  const float* b1     = (const float*)d_in[7];
  const float* W2     = (const float*)d_in[8];
  const float* b2     = (const float*)d_in[9];
  float* out = (float*)d_out;

  lds_mixture_lqr_scan<<<dim3(1), dim3(32), 0, stream>>>(
      init_y, xs1, xs2, A, gain1, gain2, W1, b1, W2, b2, out);
}